// GINEEncoder_61478161875137
// MI455X (gfx1250) — compile-verified
//
#include <hip/hip_runtime.h>

typedef __attribute__((ext_vector_type(2))) float v2f;
typedef __attribute__((ext_vector_type(8))) float v8f;

#define N_NODES   50000
#define N_EDGES   800000
#define IN_CH     64
#define EDGE_DIM  8
#define HIDDEN    128
#define NUM_GRAPHS 256

// ---------------------------------------------------------------------------
// utility: zero a float buffer
// ---------------------------------------------------------------------------
__global__ void zero_kernel(float* __restrict__ p, int n) {
    int i = blockIdx.x * blockDim.x + threadIdx.x;
    if (i < n) p[i] = 0.0f;
}

// ---------------------------------------------------------------------------
// Fused edge kernel:  agg[dst] += relu(x[src] + (edge_attr @ We + be))
// One wave (32 lanes) per edge; lane handles D/32 consecutive channels.
// We/be staged in LDS. Never materializes the [E,D] edge-feature tensor.
// ---------------------------------------------------------------------------
template<int D>
__global__ __launch_bounds__(256) void edge_msg_kernel(
    const float*     __restrict__ xin,    // [N, D]
    const long long* __restrict__ src,    // [E]
    const long long* __restrict__ dst,    // [E]
    const float*     __restrict__ eattr,  // [E, 8]
    const float*     __restrict__ We,     // [8, D]
    const float*     __restrict__ be,     // [D]
    float*           __restrict__ agg,    // [N, D]
    int E)
{
    constexpr int VEC = D / 32;           // channels per lane (2 or 4)
    __shared__ float sW[EDGE_DIM * D];
    __shared__ float sb[D];
    for (int i = threadIdx.x; i < EDGE_DIM * D; i += blockDim.x) sW[i] = We[i];
    for (int i = threadIdx.x; i < D; i += blockDim.x)            sb[i] = be[i];
    __syncthreads();

    const int lane   = threadIdx.x & 31;
    const int waveId = (blockIdx.x * blockDim.x + threadIdx.x) >> 5;
    const int nWaves = (gridDim.x * blockDim.x) >> 5;
    const int c0     = lane * VEC;

    for (long long e = waveId; e < (long long)E; e += nWaves) {
        const long long s = src[e];
        const long long d = dst[e];

        float a[EDGE_DIM];
        #pragma unroll
        for (int k = 0; k < EDGE_DIM; ++k) a[k] = eattr[e * EDGE_DIM + k];

        float m[VEC];
        #pragma unroll
        for (int v = 0; v < VEC; ++v) {
            float acc = sb[c0 + v];
            #pragma unroll
            for (int k = 0; k < EDGE_DIM; ++k)
                acc = fmaf(a[k], sW[k * D + c0 + v], acc);
            acc += xin[s * D + c0 + v];          // gather (L2-resident)
            m[v] = fmaxf(acc, 0.0f);
        }
        #pragma unroll
        for (int v = 0; v < VEC; ++v)
            atomicAdd(&agg[d * D + c0 + v], m[v]);   // global_atomic_add_f32
    }
}

// ---------------------------------------------------------------------------
// Node GEMM with WMMA:  out = relu( (X + AGG) @ W + bias )
// Block = 256 threads = 8 waves. Block owns one 16-row strip; wave w owns
// columns [16w, 16w+16). A-strip (16 x K) and full W (K x 128) staged in LDS.
// V_WMMA_F32_16X16X4_F32, K-loop in steps of 4.
// A layout (ISA 7.12.2): lanes 0-15: M=lane, {K=k,k+1}; lanes 16-31: {K=k+2,k+3}.
// D layout: vgpr r -> M = r + 8*(lane>=16), N = lane&15.
// ---------------------------------------------------------------------------
template<int K>
__global__ __launch_bounds__(256) void node_gemm_relu_kernel(
    const float* __restrict__ X,     // [M, K]
    const float* __restrict__ AGG,   // [M, K]
    const float* __restrict__ W,     // [K, HIDDEN]
    const float* __restrict__ bias,  // [HIDDEN]
    float*       __restrict__ out,   // [M, HIDDEN]
    int M)
{
    __shared__ float sA[16 * K];
    __shared__ float sW[K * HIDDEN];
    __shared__ float sb[HIDDEN];

    const int m0 = blockIdx.x * 16;

    for (int i = threadIdx.x; i < 16 * K; i += 256) {
        const int r = i / K, c = i % K;
        const long long idx = (long long)(m0 + r) * K + c;
        sA[i] = X[idx] + AGG[idx];
    }
    for (int i = threadIdx.x; i < K * HIDDEN; i += 256) sW[i] = W[i];
    for (int i = threadIdx.x; i < HIDDEN; i += 256)     sb[i] = bias[i];
    __syncthreads();

    const int wave = threadIdx.x >> 5;   // 0..7 -> column tile
    const int lane = threadIdx.x & 31;
    const int n0   = wave * 16;
    const int half = lane >> 4;          // 0 or 1
    const int lrow = lane & 15;          // M index (A) / N index (B,D)
    const int klo  = half * 2;

    v8f acc = {};
    #pragma unroll
    for (int k = 0; k < K; k += 4) {
        v2f a, b;
        a.x = sA[lrow * K + k + klo];
        a.y = sA[lrow * K + k + klo + 1];
        b.x = sW[(k + klo)     * HIDDEN + n0 + lrow];
        b.y = sW[(k + klo + 1) * HIDDEN + n0 + lrow];
        acc = __builtin_amdgcn_wmma_f32_16x16x4_f32(
                  false, a, false, b, (short)0, acc, false, false);
    }

    #pragma unroll
    for (int r = 0; r < 8; ++r) {
        const int row = m0 + r + half * 8;
        const int col = n0 + lrow;
        const float v = acc[r] + sb[col];
        out[(long long)row * HIDDEN + col] = fmaxf(v, 0.0f);
    }
}

// ---------------------------------------------------------------------------
// Global mean pool: atomic sums per graph, then divide.
// ---------------------------------------------------------------------------
__global__ void pool_sum_kernel(const float* __restrict__ x,      // [N, HIDDEN]
                                const long long* __restrict__ batch,
                                float* __restrict__ sums,          // [G, HIDDEN]
                                float* __restrict__ cnt)           // [G]
{
    const int n = blockIdx.x;
    const int c = threadIdx.x;
    const int g = (int)batch[n];
    atomicAdd(&sums[g * HIDDEN + c], x[(long long)n * HIDDEN + c]);
    if (c == 0) atomicAdd(&cnt[g], 1.0f);
}

__global__ void pool_div_kernel(float* __restrict__ sums, const float* __restrict__ cnt)
{
    const int g = blockIdx.x;
    const int c = threadIdx.x;
    sums[g * HIDDEN + c] /= fmaxf(cnt[g], 1.0f);
}

// ---------------------------------------------------------------------------
// launch
// ---------------------------------------------------------------------------
extern "C" void kernel_launch(void* const* d_in, const int* in_sizes, int n_in,
                              void* d_out, int out_size, void* d_ws, size_t ws_size,
                              hipStream_t stream)
{
    const float*     x     = (const float*)d_in[0];
    const long long* eidx  = (const long long*)d_in[1];   // [2, E]
    const float*     eattr = (const float*)d_in[2];
    const long long* batch = (const long long*)d_in[3];
    const float *W1 = (const float*)d_in[4],  *b1 = (const float*)d_in[5];
    const float *We1= (const float*)d_in[6],  *be1= (const float*)d_in[7];
    const float *W2 = (const float*)d_in[8],  *b2 = (const float*)d_in[9];
    const float *We2= (const float*)d_in[10], *be2= (const float*)d_in[11];
    const float *W3 = (const float*)d_in[12], *b3 = (const float*)d_in[13];
    const float *We3= (const float*)d_in[14], *be3= (const float*)d_in[15];

    const long long* src = eidx;
    const long long* dst = eidx + N_EDGES;

    float* ws  = (float*)d_ws;
    float* xA  = ws;                                  // [N, 128]
    float* xB  = xA + (size_t)N_NODES * HIDDEN;       // [N, 128]
    float* agg = xB + (size_t)N_NODES * HIDDEN;       // [N, 128]
    float* cnt = agg + (size_t)N_NODES * HIDDEN;      // [256]
    float* outp = (float*)d_out;                      // [256, 128]

    const int ZB = 256;
    const int EDGE_BLOCKS = 2048;                     // 8 waves/block, grid-stride
    const int GEMM_BLOCKS = N_NODES / 16;             // 3125 (exact)

    // ---- Layer 1 (IN_CH=64 -> HIDDEN=128) ----
    zero_kernel<<<(N_NODES*IN_CH + ZB-1)/ZB, ZB, 0, stream>>>(agg, N_NODES*IN_CH);
    edge_msg_kernel<IN_CH><<<EDGE_BLOCKS, 256, 0, stream>>>(
        x, src, dst, eattr, We1, be1, agg, N_EDGES);
    node_gemm_relu_kernel<IN_CH><<<GEMM_BLOCKS, 256, 0, stream>>>(
        x, agg, W1, b1, xA, N_NODES);

    // ---- Layer 2 (128 -> 128) ----
    zero_kernel<<<(N_NODES*HIDDEN + ZB-1)/ZB, ZB, 0, stream>>>(agg, N_NODES*HIDDEN);
    edge_msg_kernel<HIDDEN><<<EDGE_BLOCKS, 256, 0, stream>>>(
        xA, src, dst, eattr, We2, be2, agg, N_EDGES);
    node_gemm_relu_kernel<HIDDEN><<<GEMM_BLOCKS, 256, 0, stream>>>(
        xA, agg, W2, b2, xB, N_NODES);

    // ---- Layer 3 (128 -> 128) ----
    zero_kernel<<<(N_NODES*HIDDEN + ZB-1)/ZB, ZB, 0, stream>>>(agg, N_NODES*HIDDEN);
    edge_msg_kernel<HIDDEN><<<EDGE_BLOCKS, 256, 0, stream>>>(
        xB, src, dst, eattr, We3, be3, agg, N_EDGES);
    node_gemm_relu_kernel<HIDDEN><<<GEMM_BLOCKS, 256, 0, stream>>>(
        xB, agg, W3, b3, xA, N_NODES);

    // ---- Global mean pool ----
    zero_kernel<<<(NUM_GRAPHS*HIDDEN + ZB-1)/ZB, ZB, 0, stream>>>(outp, NUM_GRAPHS*HIDDEN);
    zero_kernel<<<1, 256, 0, stream>>>(cnt, NUM_GRAPHS);
    pool_sum_kernel<<<N_NODES, HIDDEN, 0, stream>>>(xA, batch, outp, cnt);
    pool_div_kernel<<<NUM_GRAPHS, HIDDEN, 0, stream>>>(outp, cnt);
}